// M_27109833572854
// MI455X (gfx1250) — compile-verified
//
#include <hip/hip_runtime.h>
#include <cstdint>

// ---------------------------------------------------------------------------
// RetinaFace post-process for MI455X (gfx1250, wave32).
// N = 67200 anchors, K = 1000 NMS budget, IMG = 1280.
// d_out = [boxes 1000x4 | scores 1000 | landms 1000x10] = 15000 f32.
//
// CDNA5 features used:
//   - global_load_async_to_lds_b128 + s_wait_asynccnt  (box tile staging)
//   - tensor_load_to_lds (TDM) + s_wait_tensorcnt      (125 KB mask staging)
//   - v_wmma_f32_16x16x4_f32: rank-2 union term T*(a_i+a_j+eps) per 16x16
//     IoU tile on the matrix pipe, co-executing with VALU min/max work
//   - ballot_w32 packing 32 suppression bits / instruction
// ---------------------------------------------------------------------------

#define N_ANCH   67200
#define NCHUNK   2100          // 67200 / 32
#define KTOP     1000
#define NSTRIP   63            // ceil(1000/16) row strips
#define CONF_T   0.7f
#define NMS_T    0.4f
#define IMGF     1280.0f
#define NEG_INF  (-__builtin_inff())

typedef unsigned int u32x4 __attribute__((ext_vector_type(4)));
typedef int          i32x8 __attribute__((ext_vector_type(8)));
typedef int          i32x4 __attribute__((ext_vector_type(4)));
typedef float        v2f   __attribute__((ext_vector_type(2)));
typedef float        v8f   __attribute__((ext_vector_type(8)));

// ---- kernel 1: softmax score + threshold mask ------------------------------
__global__ void k_scores(const float* __restrict__ conf, float* __restrict__ scores)
{
    int i = blockIdx.x * blockDim.x + threadIdx.x;
    if (i < N_ANCH) {
        float c0 = conf[2 * i], c1 = conf[2 * i + 1];
        float s  = __fdividef(1.0f, 1.0f + __expf(c0 - c1));   // softmax[:,1]
        scores[i] = (s > CONF_T) ? s : NEG_INF;                // thresholded key
    }
}

// ---- kernel 2a: per-32-element chunk max (one wave per chunk) --------------
__global__ void k_chunkmax(const float* __restrict__ scores, float* __restrict__ cmax)
{
    int wave = (blockIdx.x * blockDim.x + threadIdx.x) >> 5;
    int lane = threadIdx.x & 31;
    if (wave >= NCHUNK) return;
    float v = scores[wave * 32 + lane];
    for (int o = 16; o > 0; o >>= 1) v = fmaxf(v, __shfl_xor(v, o, 32));
    if (lane == 0) cmax[wave] = v;
}

// ---- kernel 2b: single-wave iterative top-K via chunk-max scan -------------
__global__ void k_topk(float* __restrict__ scores, const float* __restrict__ cmax_g,
                       int* __restrict__ top_idx, float* __restrict__ top_s)
{
    __shared__ float cmax[NCHUNK];
    int lane = threadIdx.x;                      // blockDim = 32 (one wave)
    for (int c = lane; c < NCHUNK; c += 32) cmax[c] = cmax_g[c];
    __syncthreads();

    for (int k = 0; k < KTOP; ++k) {
        float m = NEG_INF; int mc = 0;
        for (int c = lane; c < NCHUNK; c += 32) {
            float v = cmax[c];
            if (v > m) { m = v; mc = c; }
        }
        for (int o = 16; o > 0; o >>= 1) {
            float om = __shfl_xor(m, o, 32);
            int   oc = __shfl_xor(mc, o, 32);
            if (om > m || (om == m && oc < mc)) { m = om; mc = oc; }
        }
        float v = scores[mc * 32 + lane];
        unsigned bal = __builtin_amdgcn_ballot_w32(v >= m);
        int l  = __ffs(bal) - 1;
        int gi = mc * 32 + l;
        if (lane == 0) { top_idx[k] = gi; top_s[k] = m; }
        float v2 = (lane == l) ? NEG_INF : v;
        if (lane == l) scores[gi] = NEG_INF;
        for (int o = 16; o > 0; o >>= 1) v2 = fmaxf(v2, __shfl_xor(v2, o, 32));
        if (lane == 0) cmax[mc] = v2;
        __syncthreads();
    }
}

// ---- kernel 3: decode boxes + landmarks only for the K winners -------------
__global__ void k_decode(const float* __restrict__ loc, const float* __restrict__ land,
                         const float* __restrict__ priors, const int* __restrict__ top_idx,
                         float* __restrict__ boxes_k, float* __restrict__ landms_k)
{
    int k = blockIdx.x * blockDim.x + threadIdx.x;
    if (k >= KTOP) return;
    int i = top_idx[k];
    float px = priors[i * 4 + 0], py = priors[i * 4 + 1];
    float pw = priors[i * 4 + 2], ph = priors[i * 4 + 3];
    float l0 = loc[i * 4 + 0], l1 = loc[i * 4 + 1];
    float l2 = loc[i * 4 + 2], l3 = loc[i * 4 + 3];
    float cx = px + l0 * 0.1f * pw;
    float cy = py + l1 * 0.1f * ph;
    float w  = pw * __expf(l2 * 0.2f);
    float h  = ph * __expf(l3 * 0.2f);
    float x1 = cx - 0.5f * w, y1 = cy - 0.5f * h;
    boxes_k[k * 4 + 0] = x1 * IMGF;
    boxes_k[k * 4 + 1] = y1 * IMGF;
    boxes_k[k * 4 + 2] = (x1 + w) * IMGF;
    boxes_k[k * 4 + 3] = (y1 + h) * IMGF;
    for (int p = 0; p < 5; ++p) {
        float lx = land[i * 10 + 2 * p], ly = land[i * 10 + 2 * p + 1];
        landms_k[k * 10 + 2 * p]     = (px + lx * 0.1f * pw) * IMGF;
        landms_k[k * 10 + 2 * p + 1] = (py + ly * 0.1f * ph) * IMGF;
    }
}

// ---- kernel 4: K x K IoU suppression bitmask (WMMA tiles + ballot) ---------
// iou > T  <=>  inter*(1+T) > T*(a_i + a_j + eps)   (all terms non-negative).
// The RHS is a rank-2 matrix: one V_WMMA_F32_16X16X4_F32 per 16x16 tile with
//   A[M][:] = [T*a_i, T, 0, 0]   (16x4),   B[:][N] = [1, a_j+eps, 0, 0]^T (4x16)
// so the matrix pipe produces T*(a_i+a_j+eps) while the VALU computes inter.
// One wave per 16-row strip; two 16-col tiles form each 32-bit mask word.
__global__ void k_ioumask(const float* __restrict__ boxes_k, unsigned* __restrict__ supmask)
{
    __shared__ __align__(16) float lb[4096];       // 1024 boxes x 4 (padded), 16 KB
    int tid  = threadIdx.x;                        // blockDim = 256
    int lane = tid & 31;

    // zero the pad rows (1000..1023) so padded areas come out as 0
    for (int t = 4000 + tid; t < 4096; t += 256) lb[t] = 0.0f;
    // async-stage the 1000 real boxes global -> LDS
    for (int k = tid; k < KTOP; k += 256) {
        unsigned ldsa = (unsigned)(uintptr_t)(&lb[k * 4]);
        unsigned long long ga = (unsigned long long)(uintptr_t)(boxes_k + k * 4);
        asm volatile("global_load_async_to_lds_b128 %0, %1, off"
                     :: "v"(ldsa), "v"(ga) : "memory");
    }
    asm volatile("s_wait_asynccnt 0x0" ::: "memory");
    __syncthreads();

    int strip = blockIdx.x * (blockDim.x >> 5) + (tid >> 5);
    if (strip >= NSTRIP) return;                   // wave-uniform
    int i0 = strip * 16;

    // A fragment (16x4 f32): lanes 0-15 hold K=0,1 for M=lane; lanes 16-31 K=2,3 (=0)
    int mrow = i0 + (lane & 15);
    float rr1 = lb[mrow * 4 + 0], rr2 = lb[mrow * 4 + 1];
    float rr3 = lb[mrow * 4 + 2], rr4 = lb[mrow * 4 + 3];
    float ai_l = (rr3 - rr1) * (rr4 - rr2);
    v2f afrag;
    afrag.x = (lane < 16) ? (NMS_T * ai_l) : 0.0f;   // A[M][0] / A[M][2]
    afrag.y = (lane < 16) ? NMS_T : 0.0f;            // A[M][1] / A[M][3]

    // per-slot row boxes: D-matrix VGPR r -> rows i0+r (lanes 0-15), i0+r+8 (16-31)
    float rx1[8], ry1[8], rx2[8], ry2[8];
    int hi = (lane >> 4) << 3;                       // 0 or 8
#pragma unroll
    for (int r = 0; r < 8; ++r) {
        int mi = i0 + r + hi;
        rx1[r] = lb[mi * 4 + 0]; ry1[r] = lb[mi * 4 + 1];
        rx2[r] = lb[mi * 4 + 2]; ry2[r] = lb[mi * 4 + 3];
    }

    for (int wt = 0; wt < 32; ++wt) {                // one 32-bit mask word / iter
        unsigned bals[16];
#pragma unroll
        for (int half = 0; half < 2; ++half) {       // two 16-col tiles per word
            int j0 = wt * 32 + half * 16;
            int j  = j0 + (lane & 15);
            float cx1 = lb[j * 4 + 0], cy1 = lb[j * 4 + 1];
            float cx2 = lb[j * 4 + 2], cy2 = lb[j * 4 + 3];
            float aj  = (cx2 - cx1) * (cy2 - cy1);
            // B fragment (4x16 f32): lanes 0-15 row K=0 (=1), lanes 16-31 row K=1
            v2f bfrag;
            bfrag.x = (lane < 16) ? 1.0f : (aj + 1e-12f);
            bfrag.y = 0.0f;                          // rows K=2,3 are zero
            v8f cz = {0.f, 0.f, 0.f, 0.f, 0.f, 0.f, 0.f, 0.f};
            v8f U = __builtin_amdgcn_wmma_f32_16x16x4_f32(
                        false, afrag, false, bfrag, (short)0, cz, false, false);
            bool jvalid = j < KTOP;
#pragma unroll
            for (int r = 0; r < 8; ++r) {
                float xx1 = fmaxf(rx1[r], cx1), yy1 = fmaxf(ry1[r], cy1);
                float xx2 = fminf(rx2[r], cx2), yy2 = fminf(ry2[r], cy2);
                float iw  = fmaxf(xx2 - xx1, 0.0f), ih = fmaxf(yy2 - yy1, 0.0f);
                float lhs = (iw * ih) * (1.0f + NMS_T);
                bool pred = jvalid && (lhs > U[r]);
                bals[half * 8 + r] = __builtin_amdgcn_ballot_w32(pred);
            }
        }
        if (lane == 0) {
#pragma unroll
            for (int r = 0; r < 8; ++r) {
                int ilo = i0 + r, ihi2 = i0 + r + 8;
                unsigned a = bals[r], b = bals[8 + r];
                if (ilo  < KTOP) supmask[ilo  * 32 + wt] = (a & 0xFFFFu) | ((b & 0xFFFFu) << 16);
                if (ihi2 < KTOP) supmask[ihi2 * 32 + wt] = (a >> 16)     | (b & 0xFFFF0000u);
            }
        }
    }
}

// ---- kernel 5: TDM-stage mask into LDS, single-wave greedy NMS, write out --
__global__ void k_nms_write(const unsigned* __restrict__ supmask,
                            const float* __restrict__ top_s,
                            const float* __restrict__ boxes_k,
                            const float* __restrict__ landms_k,
                            float* __restrict__ out)
{
    extern __shared__ unsigned sm[];          // [32000] mask + [32] keep words
    unsigned* um = sm;
    unsigned* kw = sm + KTOP * 32;
    int tid = threadIdx.x;                    // blockDim = 1024

#if __has_builtin(__builtin_amdgcn_tensor_load_to_lds)
    if (tid < 32) {
        // D# for a 1-D 32000-dword tile: global supmask -> LDS um (125 KB)
        unsigned lds_base = (unsigned)(uintptr_t)um;
        unsigned long long ga = (unsigned long long)(uintptr_t)supmask;
        u32x4 g0 = { 1u,                                   // count=1, is_restore=0
                     lds_base,                             // lds_addr
                     (unsigned)(ga & 0xFFFFFFFFu),         // global_addr[31:0]
                     (unsigned)(((ga >> 32) & 0x01FFFFFFu) | 0x80000000u) }; // addr[56:32] | type=2
        i32x8 g1 = { (int)0x00020000u,                     // data_size=4B, no mask
                     (int)(((unsigned)(KTOP * 32) & 0xFFFFu) << 16),  // tensor_dim0 lo16
                     (int)((((unsigned)(KTOP * 32)) >> 16) | (1u << 16)), // tdim0 hi16 | tensor_dim1=1
                     (int)(((unsigned)(KTOP * 32) & 0xFFFFu) << 16),  // tile_dim0 = 32000
                     (int)1,                               // tile_dim1 = 1
                     (int)(KTOP * 32),                     // tensor_dim0_stride
                     0, 0 };
        i32x4 z4 = { 0, 0, 0, 0 };
#if defined(__clang_major__) && __clang_major__ >= 23
        i32x8 z8 = { 0, 0, 0, 0, 0, 0, 0, 0 };
        __builtin_amdgcn_tensor_load_to_lds(g0, g1, z4, z4, z8, 0);
#else
        __builtin_amdgcn_tensor_load_to_lds(g0, g1, z4, z4, 0);
#endif
        __builtin_amdgcn_s_wait_tensorcnt(0);
    }
#else
    for (int t = tid; t < KTOP * 32; t += 1024) um[t] = supmask[t];
#endif

    // initial keep = (score > CONF_T), packed 32 bits/word via wave32 ballot
    {
        bool valid = (tid < KTOP) && (top_s[tid] > CONF_T);
        unsigned b = __builtin_amdgcn_ballot_w32(valid);
        if ((tid & 31) == 0) kw[tid >> 5] = b;
    }
    __syncthreads();

    // greedy scan: one wave, lane L owns keep bits [32L, 32L+31]
    if (tid < 32) {
        unsigned keep = kw[tid];
        for (int i = 0; i < KTOP; ++i) {
            int wi = i >> 5, bi = i & 31;
            unsigned kwi = __shfl(keep, wi, 32);          // broadcast owner word
            if ((kwi >> bi) & 1u) {                       // row i survives -> suppress
                unsigned m = um[i * 32 + tid];
                unsigned above;
                if (tid < wi)      above = 0u;
                else if (tid > wi) above = 0xFFFFFFFFu;
                else               above = (bi == 31) ? 0u : (0xFFFFFFFFu << (bi + 1));
                keep &= ~(m & above);                     // only j > i suppressed
            }
        }
        kw[tid] = keep;
    }
    __syncthreads();

    // masked write-out: [boxes | scores | landms]
    int k = tid;
    if (k < KTOP) {
        bool kept = (kw[k >> 5] >> (k & 31)) & 1u;
        out[4000 + k] = kept ? top_s[k] : 0.0f;
        for (int c = 0; c < 4; ++c)  out[k * 4 + c]         = kept ? boxes_k[k * 4 + c]   : 0.0f;
        for (int c = 0; c < 10; ++c) out[5000 + k * 10 + c] = kept ? landms_k[k * 10 + c] : 0.0f;
    }
}

// ---------------------------------------------------------------------------
extern "C" void kernel_launch(void* const* d_in, const int* in_sizes, int n_in,
                              void* d_out, int out_size, void* d_ws, size_t ws_size,
                              hipStream_t stream)
{
    const float* loc    = (const float*)d_in[0];   // [1,N,4]
    const float* conf   = (const float*)d_in[1];   // [1,N,2]
    const float* land   = (const float*)d_in[2];   // [1,N,10]
    const float* priors = (const float*)d_in[3];   // [N,4]
    float* ws  = (float*)d_ws;
    float* out = (float*)d_out;

    // workspace layout (floats); supmask aliases scores (dead after top-k)
    float*    scores = ws;                          // [67200]
    float*    cmax   = ws + 67200;                  // [2100]
    int*      tidx   = (int*)(ws + 69300);          // [1000]
    float*    ts     = ws + 70300;                  // [1000]
    float*    bk     = ws + 71300;                  // [1000*4], 16B aligned
    float*    lmk    = ws + 75300;                  // [1000*10]
    unsigned* sup    = (unsigned*)ws;               // [1000*32], reuses scores

    k_scores  <<<(N_ANCH + 255) / 256, 256, 0, stream>>>(conf, scores);
    k_chunkmax<<<(NCHUNK * 32 + 255) / 256, 256, 0, stream>>>(scores, cmax);
    k_topk    <<<1, 32, 0, stream>>>(scores, cmax, tidx, ts);
    k_decode  <<<(KTOP + 255) / 256, 256, 0, stream>>>(loc, land, priors, tidx, bk, lmk);
    k_ioumask <<<(NSTRIP + 7) / 8, 256, 0, stream>>>(bk, sup);
    k_nms_write<<<1, 1024, (KTOP * 32 + 32) * sizeof(unsigned), stream>>>(sup, ts, bk, lmk, out);
}